// PathPolicyNet_3307124818436
// MI455X (gfx1250) — compile-verified
//
#include <hip/hip_runtime.h>

typedef float v2f __attribute__((ext_vector_type(2)));
typedef float v4f __attribute__((ext_vector_type(4)));
typedef float v8f __attribute__((ext_vector_type(8)));

#define P_TOT  32768
#define H      256
#define L      16
#define KRED   280      // path_emb(256) + edge(8) + scalar(16) after folding src/dst into c[]
#define APITCH 284      // A tile pitch (floats): staggers LDS banks, keeps 16B alignment
#define MTILE  64       // paths per block (4 M-subtiles of 16)
#define KC     56       // K-chunk staged in LDS (5 * 56 = 280)
#define NCHUNK 5

// ---------------------------------------------------------------------------
// Kernel 0: fold src/dst embedding contributions + b1 into a constant c[256].
// c[h] = b1[h] + sum_k src[k]*W1[k][h] + sum_k dst[k]*W1[256+k][h]
// ---------------------------------------------------------------------------
__global__ void const_head_kernel(const float* __restrict__ node_embs,
                                  const float* __restrict__ W1,
                                  const float* __restrict__ b1,
                                  const int* __restrict__ src_idx,
                                  const int* __restrict__ dst_idx,
                                  float* __restrict__ cvec) {
  const int h = threadIdx.x;               // 256 threads, one output column each
  const float* se = node_embs + (long)(*src_idx) * H;
  const float* de = node_embs + (long)(*dst_idx) * H;
  float acc = b1[h];
  for (int k = 0; k < H; ++k) acc += se[k] * W1[(long)k * H + h];
  for (int k = 0; k < H; ++k) acc += de[k] * W1[(long)(H + k) * H + h];
  cvec[h] = acc;
}

// ---------------------------------------------------------------------------
// Main kernel: per block of 64 paths
//   phase 1: ragged-mean gather -> LDS A tile [64 x 280]
//   phase 2: K-chunked WMMA f32 16x16x4 GEMM vs W1[512:792] staged in LDS
//            (each wave: 32x64 output tile, B fragments reused over 2 M-subtiles)
//   epilogue: h = relu(acc + c[n]); q = sum_n h*W2[n]  (shuffle + LDS atomic)
// ---------------------------------------------------------------------------
__global__ __launch_bounds__(256) void path_policy_kernel(
    const float* __restrict__ node_embs,
    const int*   __restrict__ path_nodes,
    const int*   __restrict__ path_lens,
    const float* __restrict__ edge_feats,
    const float* __restrict__ scal_feats,
    const float* __restrict__ W1,
    const float* __restrict__ W2,
    const float* __restrict__ b2,
    const float* __restrict__ cvec,
    float* __restrict__ out) {
  __shared__ float ldsA[MTILE][APITCH];          // 72.7 KB
  __shared__ float ldsB[(KC / 4) * H * 4];       // 57.3 KB, fragment-swizzled [kk][n][4]
  __shared__ float ldsQ[MTILE];

  const int tid  = threadIdx.x;
  const int lane = tid & 31;
  const int wid  = tid >> 5;                     // 8 waves
  const int p0   = blockIdx.x * MTILE;

  if (tid < MTILE) ldsQ[tid] = 0.0f;

  // ---- phase 1a: masked mean over gathered node embeddings ---------------
  // wave w handles paths w*8..w*8+7; each lane owns 8 contiguous H columns.
  {
    const int col = lane * 8;
    for (int j = 0; j < 8; ++j) {
      const int pi  = wid * 8 + j;
      const int p   = p0 + pi;
      const int len = path_lens[p];
      v4f a0 = {}, a1 = {};
      for (int t = 0; t < len; ++t) {
        const long idx = (long)path_nodes[p * L + t];
        const v4f* src = (const v4f*)(node_embs + idx * H + col);
        a0 += src[0];
        a1 += src[1];
      }
      const float inv = 1.0f / (float)(len > 0 ? len : 1);
      v4f* dst = (v4f*)&ldsA[pi][col];
      dst[0] = a0 * inv;
      dst[1] = a1 * inv;
    }
  }
  // ---- phase 1b: edge(8) + scalar(16) features -> A columns 256..279 -----
  for (int i = tid; i < MTILE * 24; i += 256) {
    const int pi = i / 24, c = i % 24;
    const float v = (c < 8) ? edge_feats[(long)(p0 + pi) * 8 + c]
                            : scal_feats[(long)(p0 + pi) * 16 + (c - 8)];
    ldsA[pi][256 + c] = v;
  }

  // ---- phase 2: WMMA GEMM, K = 280 in 5 chunks of 56 ---------------------
  const int mgrp  = wid >> 2;           // 0..1 -> M-subtiles {0,1} or {2,3}
  const int n0    = (wid & 3) * 64;     // 64-wide N slice of H=256
  const int lh    = lane & 15;
  const int khalf = lane >> 4;          // 0: K={0,1}, 1: K={2,3} of the 4-wide step
  const int arow0 = mgrp * 32 + lh;     // A row for M-subtile 0 of this wave
  const int arow1 = arow0 + 16;         // A row for M-subtile 1 of this wave

  v8f acc[2][4] = {};                   // [m-subtile][n-subtile]

  const float* W1s = W1 + (long)512 * H;   // rows 512..791 of W1
  for (int ch = 0; ch < NCHUNK; ++ch) {
    const int kbase = ch * KC;
    __syncthreads();                       // prev-chunk compute done / A tile ready
    // stage + swizzle B chunk: element (k,n) -> ldsB[((k>>2)*H + n)*4 + (k&3)]
    {
      const int n = tid;                   // one column per thread
      for (int kk = 0; kk < KC / 4; ++kk) {
        const long krow = kbase + kk * 4;
        v4f b;
        b[0] = W1s[(krow + 0) * H + n];
        b[1] = W1s[(krow + 1) * H + n];
        b[2] = W1s[(krow + 2) * H + n];
        b[3] = W1s[(krow + 3) * H + n];
        *(v4f*)&ldsB[(kk * H + n) * 4] = b;
      }
    }
    __syncthreads();
#pragma unroll
    for (int kk = 0; kk < KC / 4; ++kk) {
      const int k = kbase + kk * 4;
      // A fragments (16x4 f32): lane<16 -> K={k,k+1}; lane>=16 -> K={k+2,k+3}
      const v2f a0 = *(const v2f*)&ldsA[arow0][k + khalf * 2];
      const v2f a1 = *(const v2f*)&ldsA[arow1][k + khalf * 2];
      // B fragments (4x16 f32), conflict-free b64 loads from swizzled layout
      const float* bb = &ldsB[(kk * H + n0 + lh) * 4 + khalf * 2];
      const v2f b0 = *(const v2f*)(bb);
      const v2f b1 = *(const v2f*)(bb + 16 * 4);
      const v2f b2v = *(const v2f*)(bb + 32 * 4);
      const v2f b3 = *(const v2f*)(bb + 48 * 4);
      acc[0][0] = __builtin_amdgcn_wmma_f32_16x16x4_f32(false, a0, false, b0,  (short)0, acc[0][0], false, false);
      acc[0][1] = __builtin_amdgcn_wmma_f32_16x16x4_f32(false, a0, false, b1,  (short)0, acc[0][1], false, false);
      acc[0][2] = __builtin_amdgcn_wmma_f32_16x16x4_f32(false, a0, false, b2v, (short)0, acc[0][2], false, false);
      acc[0][3] = __builtin_amdgcn_wmma_f32_16x16x4_f32(false, a0, false, b3,  (short)0, acc[0][3], false, false);
      acc[1][0] = __builtin_amdgcn_wmma_f32_16x16x4_f32(false, a1, false, b0,  (short)0, acc[1][0], false, false);
      acc[1][1] = __builtin_amdgcn_wmma_f32_16x16x4_f32(false, a1, false, b1,  (short)0, acc[1][1], false, false);
      acc[1][2] = __builtin_amdgcn_wmma_f32_16x16x4_f32(false, a1, false, b2v, (short)0, acc[1][2], false, false);
      acc[1][3] = __builtin_amdgcn_wmma_f32_16x16x4_f32(false, a1, false, b3,  (short)0, acc[1][3], false, false);
    }
  }

  // ---- epilogue: relu(acc + c[n]) * W2[n], reduce over N ------------------
  float cn[4], wn[4];
#pragma unroll
  for (int s = 0; s < 4; ++s) {
    const int n = n0 + s * 16 + lh;
    cn[s] = cvec[n];
    wn[s] = W2[n];
  }
#pragma unroll
  for (int ms = 0; ms < 2; ++ms) {
#pragma unroll
    for (int r = 0; r < 8; ++r) {
      float part = 0.0f, v;
      v = acc[ms][0][r] + cn[0]; part += fmaxf(v, 0.0f) * wn[0];
      v = acc[ms][1][r] + cn[1]; part += fmaxf(v, 0.0f) * wn[1];
      v = acc[ms][2][r] + cn[2]; part += fmaxf(v, 0.0f) * wn[2];
      v = acc[ms][3][r] + cn[3]; part += fmaxf(v, 0.0f) * wn[3];
      // reduce across the 16 lanes of each half (rows r and r+8 live in halves)
      part += __shfl_xor(part, 1, 32);
      part += __shfl_xor(part, 2, 32);
      part += __shfl_xor(part, 4, 32);
      part += __shfl_xor(part, 8, 32);
      if (lh == 0) {
        const int row = mgrp * 32 + ms * 16 + r + khalf * 8;
        atomicAdd(&ldsQ[row], part);     // ds_add_f32 across the 4 N-slice waves
      }
    }
  }

  __syncthreads();
  if (tid < MTILE) {
    const int p   = p0 + tid;
    const int len = path_lens[p];
    out[p] = (len > 0) ? (ldsQ[tid] + b2[0]) : 0.0f;
  }
}

// ---------------------------------------------------------------------------
extern "C" void kernel_launch(void* const* d_in, const int* in_sizes, int n_in,
                              void* d_out, int out_size, void* d_ws, size_t ws_size,
                              hipStream_t stream) {
  const float* node_embs = (const float*)d_in[0];
  const int*   path_nodes = (const int*)d_in[1];
  const int*   path_lens  = (const int*)d_in[2];
  const float* edge_feats = (const float*)d_in[3];
  const float* scal_feats = (const float*)d_in[4];
  const float* W1 = (const float*)d_in[5];
  const float* b1 = (const float*)d_in[6];
  const float* W2 = (const float*)d_in[7];
  const float* b2 = (const float*)d_in[8];
  const int*   src_idx = (const int*)d_in[9];
  const int*   dst_idx = (const int*)d_in[10];

  float* cvec = (float*)d_ws;   // 256 floats of scratch

  const_head_kernel<<<1, 256, 0, stream>>>(node_embs, W1, b1, src_idx, dst_idx, cvec);
  path_policy_kernel<<<P_TOT / MTILE, 256, 0, stream>>>(
      node_embs, path_nodes, path_lens, edge_feats, scal_feats,
      W1, W2, b2, cvec, (float*)d_out);
}